// TemporalRGCNLayer_28226525069512
// MI455X (gfx1250) — compile-verified
//
#include <hip/hip_runtime.h>
#include <hip/hip_bf16.h>

// ---------------- problem constants (from reference) ----------------
#define D    128
#define R    16
#define NB   10
#define KAGG (R * D)        // 2048
#define KTOT (KAGG + D + D) // 2304  = [agg | x | x_time]
#define NCHUNK (KTOT / 64)  // 36
#define DECAY_RATE 0.1f
#define LN_EPS 1e-5f

typedef float v2f __attribute__((ext_vector_type(2)));
typedef float v8f __attribute__((ext_vector_type(8)));

// ---- gfx1250 async global->LDS path (guarded; falls back to float4 copy) ----
#if defined(__gfx1250__) && __has_builtin(__builtin_amdgcn_global_load_async_to_lds_b128) && \
    __has_builtin(__builtin_amdgcn_s_wait_asynccnt)
#define ASYNC_LDS 1
typedef int v4i_ __attribute__((vector_size(16)));               // matches builtin param pointee
typedef __attribute__((address_space(1))) v4i_ as1_v4i;          // global source
typedef __attribute__((address_space(3))) v4i_ as3_v4i;          // LDS destination
#else
#define ASYNC_LDS 0
#endif

__device__ __forceinline__ void cp_b128_to_lds(const float* gsrc, float* ldst) {
#if ASYNC_LDS
    __builtin_amdgcn_global_load_async_to_lds_b128(
        (as1_v4i*)gsrc, (as3_v4i*)ldst, 0, 0);
#else
    *(float4*)ldst = *(const float4*)gsrc;
#endif
}

__device__ __forceinline__ void async_copies_wait() {
#if ASYNC_LDS
    __builtin_amdgcn_s_wait_asynccnt(0);
#endif
}

// B_cat is stored K-pair interleaved: element (k, o) lives at flat index
//   (k & ~1) * 128 + o * 2 + (k & 1)
// so that the two values a WMMA K=4 step needs per lane -- B[k][o], B[k+1][o]
// (k even) -- are adjacent 8 bytes in LDS => single ds_load_b64, no repacking.
__device__ __forceinline__ size_t bswz(int k, int o) {
    return (size_t)(k & ~1) * D + o * 2 + (k & 1);
}

// ---------------- kernel 0: init scalar slots ----------------
__global__ void init_scalars(unsigned* maxb, unsigned* minb, float* wsum) {
    if (threadIdx.x == 0) {
        *maxb = 0u;          // edge_time >= 0, bits monotonic
        *minb = 0x7F7FFFFFu; // +FLT_MAX bits
        *wsum = 0.0f;
    }
}

// ---------------- kernel 1: build B_cat[KTOT][D] (pair-swizzled) ----------------
// rows 0..2047   : W_cat[(r*128+d)][o] = sum_b comp[r,b] * basis[b,d,o]
// rows 2048..2175: root[d][o]
// rows 2176..2303: tp_w[d][o]
__global__ void build_B(const float* __restrict__ basis, const float* __restrict__ comp,
                        const float* __restrict__ root, const float* __restrict__ tp_w,
                        float* __restrict__ Bcat) {
    int idx = blockIdx.x * blockDim.x + threadIdx.x; // k*128 + o
    if (idx >= KTOT * D) return;
    int k = idx >> 7;
    int o = idx & (D - 1);
    float v;
    if (k < KAGG) {
        int r = k >> 7;        // relation
        int d = k & (D - 1);   // input dim
        float acc = 0.0f;
        #pragma unroll
        for (int b = 0; b < NB; ++b)
            acc += comp[r * NB + b] * basis[((size_t)b * D + d) * D + o];
        v = acc;
    } else if (k < KAGG + D) {
        v = root[(size_t)(k - KAGG) * D + o];
    } else {
        v = tp_w[(size_t)(k - KAGG - D) * D + o];
    }
    Bcat[bswz(k, o)] = v;
}

// ---------------- kernel 2: min/max of edge_time (uint-bit trick, times >= 0) --------
__global__ void time_minmax(const float* __restrict__ et, int E,
                            unsigned* __restrict__ maxb, unsigned* __restrict__ minb) {
    __shared__ unsigned smax[256], smin[256];
    int tid = threadIdx.x;
    int i = blockIdx.x * 256 + tid;
    unsigned u_max = 0u, u_min = 0x7F7FFFFFu;
    if (i < E) {
        unsigned u = __float_as_uint(et[i]);
        u_max = u; u_min = u;
    }
    smax[tid] = u_max; smin[tid] = u_min;
    __syncthreads();
    for (int off = 128; off > 0; off >>= 1) {
        if (tid < off) {
            smax[tid] = max(smax[tid], smax[tid + off]);
            smin[tid] = min(smin[tid], smin[tid + off]);
        }
        __syncthreads();
    }
    if (tid == 0) {
        atomicMax(maxb, smax[0]);
        atomicMin(minb, smin[0]);
    }
}

// ---------------- kernel 3: sum of exp(-rate * td_norm) ----------------
__global__ void time_wsum(const float* __restrict__ et, int E,
                          const unsigned* __restrict__ maxb, const unsigned* __restrict__ minb,
                          float* __restrict__ wsum) {
    __shared__ float ssum[256];
    int tid = threadIdx.x;
    int i = blockIdx.x * 256 + tid;
    float maxet = __uint_as_float(*maxb);
    float span  = maxet - __uint_as_float(*minb); // td.max() == max - min
    float inv_span = 1.0f / (span + 1e-8f);
    float w = 0.0f;
    if (i < E) {
        float td = (maxet - et[i]) * inv_span;
        w = __expf(-DECAY_RATE * td);
    }
    ssum[tid] = w;
    __syncthreads();
    for (int off = 128; off > 0; off >>= 1) {
        if (tid < off) ssum[tid] += ssum[tid + off];
        __syncthreads();
    }
    if (tid == 0) atomicAdd(wsum, ssum[0]);
}

// ---------------- kernel 4: edge scatter (one wave32 per edge) ----------------
__global__ void edge_scatter(const float* __restrict__ x,
                             const int* __restrict__ ei,       // [2,E]
                             const int* __restrict__ etype,
                             const float* __restrict__ et, int E,
                             const unsigned* __restrict__ maxb, const unsigned* __restrict__ minb,
                             const float* __restrict__ wsum,
                             float* __restrict__ agg,   // [N*R, D]
                             float* __restrict__ cnt,   // [N*R]
                             float* __restrict__ aggt,  // [N, D]
                             float* __restrict__ deg)   // [N]
{
    int lane = threadIdx.x & 31;
    int e = blockIdx.x * 8 + (threadIdx.x >> 5);
    if (e >= E) return;
    int src = ei[e];
    int dst = ei[E + e];
    int rel = etype[e];
    float maxet = __uint_as_float(*maxb);
    float span  = maxet - __uint_as_float(*minb);
    float td = (maxet - et[e]) / (span + 1e-8f);
    float we = __expf(-DECAY_RATE * td) / (*wsum + 1e-8f);

    const float4* xv = (const float4*)(x + (size_t)src * D);
    float4 v = xv[lane]; // lane owns dims [4*lane, 4*lane+4)

    float* pa = agg  + ((size_t)dst * R + rel) * D + lane * 4;
    float* pt = aggt + (size_t)dst * D + lane * 4;
    atomicAdd(pa + 0, v.x);  atomicAdd(pa + 1, v.y);
    atomicAdd(pa + 2, v.z);  atomicAdd(pa + 3, v.w);
    atomicAdd(pt + 0, v.x * we);  atomicAdd(pt + 1, v.y * we);
    atomicAdd(pt + 2, v.z * we);  atomicAdd(pt + 3, v.w * we);
    if (lane == 0) {
        atomicAdd(&cnt[(size_t)dst * R + rel], 1.0f);
        atomicAdd(&deg[dst], 1.0f);
    }
}

// ---------------- kernel 5: reciprocals ----------------
__global__ void recips(const float* __restrict__ cnt, const float* __restrict__ deg,
                       float* __restrict__ invcnt, float* __restrict__ invdeg, int NR, int N) {
    int i = blockIdx.x * blockDim.x + threadIdx.x;
    if (i < NR) invcnt[i] = 1.0f / fmaxf(cnt[i], 1.0f);
    if (i < N)  invdeg[i] = 1.0f / fmaxf(deg[i], 1.0f);
}

// ---------------- kernel 6: fused WMMA GEMM + bias + ReLU + LayerNorm -------------
// Block = 256 threads = 8 wave32. Block computes rows [16*bid, 16*bid+16) x all 128 cols.
// Wave w owns cols [16w, 16w+16). K = 2304 in double-buffered LDS chunks of 64:
//   B chunk (64x128 = 32KB, pair-swizzled, pure copy) -> async global->LDS DMA (ASYNCcnt)
//   A chunk (16x64, mean-normalized)                  -> VALU staging (scale folded in)
// Chunk k+1's staging overlaps chunk k's 16 WMMA steps.
__global__ __launch_bounds__(256) void gemm_ln(
        const float* __restrict__ Bcat,    // [KTOT, D] pair-swizzled
        const float* __restrict__ agg,     // [N*R, D]
        const float* __restrict__ invcnt,  // [N*R]
        const float* __restrict__ x,       // [N, D]
        const float* __restrict__ aggt,    // [N, D]
        const float* __restrict__ invdeg,  // [N]
        const float* __restrict__ bias, const float* __restrict__ tp_b,
        const float* __restrict__ gamma, const float* __restrict__ beta,
        float* __restrict__ out)
{
    __shared__ __align__(16) float As[2][16 * 64];    //  2 x 4 KB
    __shared__ __align__(16) float Bs[2][64 * 128];   //  2 x 32 KB (swizzled chunk)
    __shared__ __align__(16) float Hs[16 * 128];      //  8 KB post-ReLU tile
    __shared__ float rs[256], rss[256];

    const int tid  = threadIdx.x;
    const int wave = tid >> 5;
    const int lane = tid & 31;
    const int m0   = blockIdx.x * 16;

    const int mrow = lane & 15;          // A row within tile
    const int koff = (lane >> 4) * 2;    // K pair: lanes 0-15 -> {0,1}, lanes 16-31 -> {2,3}
    const int col  = wave * 16 + (lane & 15);

    // stage A chunk (with on-the-fly mean normalization) into As[buf]
    auto stage_A = [&](int chunk, int buf) {
        const int kb = chunk * 64;
        for (int i = tid; i < 16 * 64; i += 256) {
            int m  = i >> 6;
            int kk = i & 63;
            int k  = kb + kk;
            int n  = m0 + m;
            float v;
            if (k < KAGG) {
                int r = k >> 7;
                v = agg[((size_t)n * R + r) * D + (k & (D - 1))] * invcnt[(size_t)n * R + r];
            } else if (k < KAGG + D) {
                v = x[(size_t)n * D + (k - KAGG)];
            } else {
                v = aggt[(size_t)n * D + (k - KAGG - D)] * invdeg[n];
            }
            As[buf][m * 64 + kk] = v;
        }
    };
    // issue async DMA of B chunk (64x128 floats, contiguous even when swizzled) into Bs[buf]
    auto issue_B = [&](int chunk, int buf) {
        const float* Bg = Bcat + (size_t)chunk * 64 * D;
        for (int i = tid; i < 64 * 128 / 4; i += 256)
            cp_b128_to_lds(Bg + i * 4, &Bs[buf][i * 4]);
    };

    v8f c = {};

    issue_B(0, 0);
    stage_A(0, 0);
    async_copies_wait();
    __syncthreads();

    int cur = 0;
    for (int chunk = 0; chunk < NCHUNK; ++chunk) {
        const int nxt = cur ^ 1;
        if (chunk + 1 < NCHUNK) {       // overlap next chunk's staging with this chunk's WMMA
            issue_B(chunk + 1, nxt);
            stage_A(chunk + 1, nxt);
        }
        // ---- 16 WMMA steps of K=4 each on the current buffers ----
        #pragma unroll
        for (int s = 0; s < 16; ++s) {
            int k = s * 4 + koff;                             // k even
            v2f a = *(const v2f*)&As[cur][mrow * 64 + k];     // A[m][k], A[m][k+1]
            v2f b = *(const v2f*)&Bs[cur][(size_t)k * 128 + col * 2]; // B[k][col], B[k+1][col]
            c = __builtin_amdgcn_wmma_f32_16x16x4_f32(
                    false, a, false, b, (short)0, c, false, false);
        }
        async_copies_wait();
        __syncthreads();
        cur = nxt;
    }

    // ---- epilogue: + (bias + tp_b), ReLU, stash tile in LDS ----
    {
        float add = bias[col] + tp_b[col];
        int mbase = (lane >> 4) * 8;     // C layout: VGPR j holds M=j (lanes 0-15) / M=j+8 (16-31)
        #pragma unroll
        for (int j = 0; j < 8; ++j) {
            float v = c[j] + add;
            Hs[(mbase + j) * 128 + col] = v > 0.0f ? v : 0.0f;
        }
    }
    __syncthreads();

    // ---- row LayerNorm: 16 threads per row, 8 cols each ----
    const int row = tid >> 4;
    const int sub = tid & 15;
    float s = 0.0f, ss = 0.0f;
    for (int ci = sub; ci < 128; ci += 16) {
        float v = Hs[row * 128 + ci];
        s += v; ss += v * v;
    }
    rs[tid] = s; rss[tid] = ss;
    __syncthreads();
    for (int off = 8; off > 0; off >>= 1) {
        if (sub < off) { rs[tid] += rs[tid + off]; rss[tid] += rss[tid + off]; }
        __syncthreads();
    }
    float mu   = rs[row * 16] * (1.0f / 128.0f);
    float var  = rss[row * 16] * (1.0f / 128.0f) - mu * mu;
    float rstd = rsqrtf(var + LN_EPS);
    for (int ci = sub; ci < 128; ci += 16) {
        float v = Hs[row * 128 + ci];
        out[((size_t)(m0 + row)) * 128 + ci] = (v - mu) * rstd * gamma[ci] + beta[ci];
    }
}

// ---------------- host launcher ----------------
extern "C" void kernel_launch(void* const* d_in, const int* in_sizes, int n_in,
                              void* d_out, int out_size, void* d_ws, size_t ws_size,
                              hipStream_t stream) {
    const float* x     = (const float*)d_in[0];
    const int*   ei    = (const int*)  d_in[1];  // [2, E]
    const int*   etype = (const int*)  d_in[2];
    const float* et    = (const float*)d_in[3];
    const float* basis = (const float*)d_in[4];
    const float* comp  = (const float*)d_in[5];
    const float* root  = (const float*)d_in[6];
    const float* bias  = (const float*)d_in[7];
    const float* tp_w  = (const float*)d_in[8];
    const float* tp_b  = (const float*)d_in[9];
    const float* gamma = (const float*)d_in[10];
    const float* beta  = (const float*)d_in[11];
    float* out = (float*)d_out;

    const int N  = in_sizes[0] / D;
    const int E  = in_sizes[2];
    const int NR = N * R;

    // ---- workspace layout (floats) ----
    float*    w      = (float*)d_ws;
    float*    Bcat   = w;                        // KTOT*D
    float*    invcnt = Bcat + (size_t)KTOT * D;  // NR
    float*    invdeg = invcnt + NR;              // N
    unsigned* maxb   = (unsigned*)(invdeg + N);  // 1
    unsigned* minb   = maxb + 1;                 // 1
    float*    wsum   = (float*)(minb + 1);       // 1
    float*    cnt    = wsum + 1;                 // NR
    float*    deg    = cnt + NR;                 // N
    float*    aggt   = deg + N;                  // N*D
    float*    agg    = aggt + (size_t)N * D;     // NR*D

    // zero accumulators (cnt, deg, aggt, agg are contiguous)
    size_t zbytes = ((size_t)NR + N + (size_t)N * D + (size_t)NR * D) * sizeof(float);
    (void)hipMemsetAsync(cnt, 0, zbytes, stream);
    init_scalars<<<1, 32, 0, stream>>>(maxb, minb, wsum);

    build_B<<<(KTOT * D + 255) / 256, 256, 0, stream>>>(basis, comp, root, tp_w, Bcat);

    int rblocks = (E + 255) / 256;
    time_minmax<<<rblocks, 256, 0, stream>>>(et, E, maxb, minb);
    time_wsum<<<rblocks, 256, 0, stream>>>(et, E, maxb, minb, wsum);

    edge_scatter<<<(E + 7) / 8, 256, 0, stream>>>(x, ei, etype, et, E, maxb, minb, wsum,
                                                  agg, cnt, aggt, deg);

    recips<<<(NR + 255) / 256, 256, 0, stream>>>(cnt, deg, invcnt, invdeg, NR, N);

    gemm_ln<<<N / 16, 256, 0, stream>>>(Bcat, agg, invcnt, x, aggt, invdeg,
                                        bias, tp_b, gamma, beta, out);
}